// RoIHeadTemplate_15350213116278
// MI455X (gfx1250) — compile-verified
//
#include <hip/hip_runtime.h>
#include <hip/hip_bf16.h>
#include <math.h>

// ---------------- problem constants (match reference) ----------------
#define BATCH      4
#define NPTS       8192
#define PRE        512          // NMS_PRE == FT_PRE
#define NMSPOST    128
#define FTPOST     64
#define NMS_TH     0.7f
#define FT_TH      0.85f
#define MASK_TH    0.3f
#define LAMB       0.5f
#define EPSF       1e-8f

// output layout (floats): rois[B*256*7] | scores[B*256] | labels[B*256] | len[B]
#define OUT_ROIS   0
#define OUT_SCORES (BATCH*256*7)
#define OUT_LABELS (BATCH*256*7 + BATCH*256)
#define OUT_LEN    (BATCH*256*7 + 2*BATCH*256)

typedef __attribute__((ext_vector_type(2))) float v2f;
typedef __attribute__((ext_vector_type(8))) float v8f;

// =====================================================================
// device helpers: rotated box geometry
// =====================================================================
__device__ inline void box_corners(const float* b, float* cx, float* cy) {
    float co = cosf(b[6]), si = sinf(b[6]);
    float hx = 0.5f * b[3], hy = 0.5f * b[4];
    const float sx[4] = { 1.f, -1.f, -1.f,  1.f };
    const float sy[4] = { 1.f,  1.f, -1.f, -1.f };
#pragma unroll
    for (int k = 0; k < 4; ++k) {
        float lx = sx[k] * hx, ly = sy[k] * hy;
        cx[k] = lx * co - ly * si + b[0];
        cy[k] = lx * si + ly * co + b[1];
    }
}

// Sutherland-Hodgman convex quad intersection area (both quads CCW)
__device__ float quad_inter_area(const float* a, const float* b) {
    float ax[4], ay[4], bx[4], by[4];
    box_corners(a, ax, ay);
    box_corners(b, bx, by);
    float px[16], py[16], qx[16], qy[16];
    int n = 4;
#pragma unroll
    for (int k = 0; k < 4; ++k) { px[k] = ax[k]; py[k] = ay[k]; }
    for (int e = 0; e < 4 && n > 0; ++e) {
        float x1 = bx[e], y1 = by[e];
        float x2 = bx[(e + 1) & 3], y2 = by[(e + 1) & 3];
        float ex = x2 - x1, ey = y2 - y1;
        int m = 0;
        for (int v = 0; v < n; ++v) {
            int w = (v + 1 == n) ? 0 : v + 1;
            float cx0 = px[v], cy0 = py[v];
            float cx1 = px[w], cy1 = py[w];
            float d0 = ex * (cy0 - y1) - ey * (cx0 - x1);
            float d1 = ex * (cy1 - y1) - ey * (cx1 - x1);
            bool in0 = d0 >= -1e-6f, in1 = d1 >= -1e-6f;
            if (in0) { qx[m] = cx0; qy[m] = cy0; ++m; }
            if (in0 != in1) {
                float t = d0 / (d0 - d1);
                qx[m] = cx0 + t * (cx1 - cx0);
                qy[m] = cy0 + t * (cy1 - cy0);
                ++m;
            }
        }
        n = m;
        for (int v = 0; v < n; ++v) { px[v] = qx[v]; py[v] = qy[v]; }
    }
    if (n < 3) return 0.f;
    float area = 0.f;
    for (int v = 0; v < n; ++v) {
        int w = (v + 1 == n) ? 0 : v + 1;
        area += px[v] * py[w] - px[w] * py[v];
    }
    return 0.5f * fabsf(area);
}

// =====================================================================
// K0: zero the output buffer
// =====================================================================
__global__ void k_zero(float* out, int n) {
    int i = blockIdx.x * blockDim.x + threadIdx.x;
    if (i < n) out[i] = 0.f;
}

// =====================================================================
// K1: score = max(cls), label = argmax(cls) over 2 classes
// =====================================================================
__global__ void k_score_label(const float* __restrict__ cls,
                              float* __restrict__ scores,
                              int* __restrict__ labels, int total) {
    int i = blockIdx.x * blockDim.x + threadIdx.x;
    if (i >= total) return;
    float c0 = cls[2 * i], c1 = cls[2 * i + 1];
    scores[i] = fmaxf(c0, c1);
    labels[i] = (c1 > c0) ? 1 : 0;     // argmax picks first on tie
}

// =====================================================================
// K2: per-batch bitonic top-512: full 8192-key sort in LDS (64KB)
//     then emit top boxes + nx (|c|^2) + rad (half-diagonal) for gate
// =====================================================================
__global__ __launch_bounds__(1024) void k_topk(
    const float* __restrict__ scores, const float* __restrict__ boxes,
    int* __restrict__ top_idx, float* __restrict__ top_scores,
    float* __restrict__ top_boxes, float* __restrict__ nx2,
    float* __restrict__ rad) {
    __shared__ float skey[NPTS];
    __shared__ int   sidx[NPTS];
    const int b = blockIdx.x;
    const float* sc = scores + b * NPTS;
    for (int t = threadIdx.x; t < NPTS; t += blockDim.x) {
        skey[t] = sc[t];
        sidx[t] = t;
    }
    // bitonic sort, descending by (score desc, idx asc)
    for (int k = 2; k <= NPTS; k <<= 1) {
        for (int j = k >> 1; j > 0; j >>= 1) {
            __syncthreads();
            for (int t = threadIdx.x; t < NPTS; t += blockDim.x) {
                int ixj = t ^ j;
                if (ixj > t) {
                    float ka = skey[t], kb = skey[ixj];
                    int ia = sidx[t], ib = sidx[ixj];
                    bool a_first = (ka > kb) || (ka == kb && ia < ib);
                    bool ddir = ((t & k) == 0);       // descending segment
                    bool sw = ddir ? (!a_first) : a_first;
                    if (sw) {
                        skey[t] = kb; skey[ixj] = ka;
                        sidx[t] = ib; sidx[ixj] = ia;
                    }
                }
            }
        }
    }
    __syncthreads();
    if (threadIdx.x < PRE) {
        int t = threadIdx.x;
        int gi = sidx[t];
        top_idx[b * PRE + t] = gi;
        top_scores[b * PRE + t] = skey[t];
        const float* src = boxes + (size_t)b * NPTS * 7 + (size_t)gi * 7;
        float* dst = top_boxes + ((size_t)b * PRE + t) * 7;
#pragma unroll
        for (int c = 0; c < 7; ++c) dst[c] = src[c];
        float x = src[0], y = src[1], dx = src[3], dy = src[4];
        nx2[b * PRE + t] = x * x + y * y;
        rad[b * PRE + t] = 0.5f * sqrtf(dx * dx + dy * dy);
    }
}

// =====================================================================
// K3: WMMA F32 16x16x4 center dot-product tiles -> overlap gate.
//     One wave32 per 16x16 tile of the 512x512 matrix; 8 waves/block.
//     A = centers[m] x [x y 0 0], B = [x y 0 0]^T x centers[n]:
//     f32 A 16x4 layout: lanes 0-15 carry K=0,1 in v[0],v[1]; hi lanes K=2,3=0
// =====================================================================
__global__ __launch_bounds__(256) void k_gate_wmma(
    const float* __restrict__ top_boxes, const float* __restrict__ nx2,
    const float* __restrict__ rad, unsigned char* __restrict__ gate) {
    const int b = blockIdx.y;
    const int wave = threadIdx.x >> 5;
    const int lane = threadIdx.x & 31;
    const int tile = blockIdx.x * 8 + wave;       // 0..1023
    const int ti = tile >> 5, tj = tile & 31;     // 32x32 tiles of 16
    const bool hi = lane >= 16;
    const int lm = lane & 15;

    const float* tb = top_boxes + (size_t)b * PRE * 7;
    const float* nn = nx2 + b * PRE;
    const float* rr = rad + b * PRE;

    int m = ti * 16 + lm;
    int n = tj * 16 + lm;
    v2f a, bb;
    a.x  = hi ? 0.f : tb[m * 7 + 0];   // K=0 (x) for lanes 0-15, K=2 pad
    a.y  = hi ? 0.f : tb[m * 7 + 1];   // K=1 (y)            , K=3 pad
    bb.x = hi ? 0.f : tb[n * 7 + 0];
    bb.y = hi ? 0.f : tb[n * 7 + 1];

    v8f acc = {};
    acc = __builtin_amdgcn_wmma_f32_16x16x4_f32(
        false, a, false, bb, (short)0, acc, false, false);

    unsigned char* g = gate + (size_t)b * PRE * PRE;
#pragma unroll
    for (int r = 0; r < 8; ++r) {
        int mm = ti * 16 + r + (hi ? 8 : 0);
        int nc = tj * 16 + lm;
        float dot = acc[r];
        float d2 = nn[mm] + nn[nc] - 2.f * dot;    // |c_m - c_n|^2
        float s = rr[mm] + rr[nc];
        g[(size_t)mm * PRE + nc] = (d2 <= s * s + 1.0f) ? 1 : 0;
    }
}

// =====================================================================
// K4: gated 512x512 rotated BEV IoU matrix (thread per pair)
// =====================================================================
__global__ __launch_bounds__(256) void k_iou_bev(
    const float* __restrict__ top_boxes, const unsigned char* __restrict__ gate,
    float* __restrict__ iou) {
    const int b = blockIdx.y;
    int p = blockIdx.x * blockDim.x + threadIdx.x;
    if (p >= PRE * PRE) return;
    int i = p / PRE, j = p % PRE;
    float* out = iou + (size_t)b * PRE * PRE + p;
    if (!gate[(size_t)b * PRE * PRE + p]) { *out = 0.f; return; }
    const float* ba = top_boxes + ((size_t)b * PRE + i) * 7;
    const float* bbx = top_boxes + ((size_t)b * PRE + j) * 7;
    float inter = quad_inter_area(ba, bbx);
    float aa = ba[3] * ba[4], ab = bbx[3] * bbx[4];
    *out = inter / fmaxf(aa + ab - inter, EPSF);
}

// =====================================================================
// K5: sequential dual-threshold NMS scan + ordered selection
// =====================================================================
__global__ __launch_bounds__(512) void k_nms_scan(
    const float* __restrict__ iou,
    int* __restrict__ nms_sel, int* __restrict__ nms_val,
    int* __restrict__ ft_sel, int* __restrict__ ft_val) {
    const int b = blockIdx.x;
    const float* m = iou + (size_t)b * PRE * PRE;
    __shared__ unsigned char sup_n[PRE], sup_f[PRE];
    int j = threadIdx.x;
    sup_n[j] = 0; sup_f[j] = 0;
    for (int i = 0; i < PRE; ++i) {
        __syncthreads();
        bool ln = !sup_n[i], lf = !sup_f[i];
        __builtin_prefetch(&m[(size_t)(i + 1) * PRE + j], 0, 1);
        if (j > i) {
            float v = m[(size_t)i * PRE + j];
            if (ln && v > NMS_TH) sup_n[j] = 1;
            if (lf && v > FT_TH)  sup_f[j] = 1;
        }
    }
    __syncthreads();
    if (threadIdx.x == 0) {
        // order = kept (idx asc) then suppressed (idx asc); take first POST
        int c = 0;
        for (int i = 0; i < PRE && c < NMSPOST; ++i)
            if (!sup_n[i]) { nms_sel[b * NMSPOST + c] = i; nms_val[b * NMSPOST + c] = 1; ++c; }
        for (int i = 0; i < PRE && c < NMSPOST; ++i)
            if (sup_n[i]) { nms_sel[b * NMSPOST + c] = i; nms_val[b * NMSPOST + c] = 0; ++c; }
        c = 0;
        for (int i = 0; i < PRE && c < FTPOST; ++i)
            if (!sup_f[i]) { ft_sel[b * FTPOST + c] = i; ft_val[b * FTPOST + c] = 1; ++c; }
        for (int i = 0; i < PRE && c < FTPOST; ++i)
            if (sup_f[i]) { ft_sel[b * FTPOST + c] = i; ft_val[b * FTPOST + c] = 0; ++c; }
    }
}

// =====================================================================
// K6: write coarse rois/scores/labels to d_out; stage s_* / f_* arrays
// =====================================================================
__global__ __launch_bounds__(128) void k_select(
    const float* __restrict__ top_boxes, const float* __restrict__ top_scores,
    const int* __restrict__ top_idx, const int* __restrict__ labels,
    const int* __restrict__ nms_sel, const int* __restrict__ nms_val,
    const int* __restrict__ ft_sel,
    float* __restrict__ out,
    float* __restrict__ sboxes, int* __restrict__ slabels,
    float* __restrict__ fboxes, float* __restrict__ fscores,
    int* __restrict__ flabels) {
    const int b = blockIdx.x;
    const int t = threadIdx.x;
    if (t < NMSPOST) {
        int sel = nms_sel[b * NMSPOST + t];
        int val = nms_val[b * NMSPOST + t];
        const float* bx = top_boxes + ((size_t)b * PRE + sel) * 7;
        int gi = top_idx[b * PRE + sel];
        int lab = labels[b * NPTS + gi];
        float sc = top_scores[b * PRE + sel];
        float* ro = out + OUT_ROIS + ((size_t)b * 256 + t) * 7;
#pragma unroll
        for (int c = 0; c < 7; ++c) ro[c] = val ? bx[c] : 0.f;
        out[OUT_SCORES + b * 256 + t] = val ? sc : 0.f;
        out[OUT_LABELS + b * 256 + t] = (float)((val ? lab : 0) + 1);
        // stage raw s_boxes / s_labels for the fine stage
        float* sb = sboxes + ((size_t)b * NMSPOST + t) * 7;
#pragma unroll
        for (int c = 0; c < 7; ++c) sb[c] = bx[c];
        slabels[b * NMSPOST + t] = lab;
    }
    if (t < FTPOST) {
        int sel = ft_sel[b * FTPOST + t];
        const float* bx = top_boxes + ((size_t)b * PRE + sel) * 7;
        float* fb = fboxes + ((size_t)b * FTPOST + t) * 7;
#pragma unroll
        for (int c = 0; c < 7; ++c) fb[c] = bx[c];
        fscores[b * FTPOST + t] = top_scores[b * PRE + sel];
        flabels[b * FTPOST + t] = labels[b * NPTS + top_idx[b * PRE + sel]];
    }
}

// =====================================================================
// K7: 64x128 IoU3D matrix (f_boxes x s_boxes)
// =====================================================================
__global__ __launch_bounds__(256) void k_iou3d(
    const float* __restrict__ fboxes, const float* __restrict__ sboxes,
    float* __restrict__ iou3) {
    const int b = blockIdx.y;
    int p = blockIdx.x * blockDim.x + threadIdx.x;
    if (p >= FTPOST * NMSPOST) return;
    int f = p / NMSPOST, s = p % NMSPOST;
    const float* ba = fboxes + ((size_t)b * FTPOST + f) * 7;
    const float* bb = sboxes + ((size_t)b * NMSPOST + s) * 7;
    float inter_bev = quad_inter_area(ba, bb);
    float top = fminf(ba[2] + 0.5f * ba[5], bb[2] + 0.5f * bb[5]);
    float bot = fmaxf(ba[2] - 0.5f * ba[5], bb[2] - 0.5f * bb[5]);
    float h = fmaxf(top - bot, 0.f);
    float inter = inter_bev * h;
    float va = ba[3] * ba[4] * ba[5], vb = bb[3] * bb[4] * bb[5];
    iou3[(size_t)b * FTPOST * NMSPOST + p] =
        inter / fmaxf(va + vb - inter, EPSF);
}

// =====================================================================
// K8: per-class top-2 match, merge, compacting scatter, len_out
// =====================================================================
__global__ __launch_bounds__(64) void k_fine(
    const float* __restrict__ iou3,
    const float* __restrict__ fboxes, const float* __restrict__ fscores,
    const int* __restrict__ flabels, const int* __restrict__ ft_val,
    const float* __restrict__ sboxes, const int* __restrict__ slabels,
    const int* __restrict__ nms_val,
    float* __restrict__ out) {
    const int b = blockIdx.x;
    const int f = threadIdx.x;          // 64 threads
    __shared__ unsigned char m0[FTPOST], m1[FTPOST];
    __shared__ int i0[FTPOST], i1[FTPOST];
    __shared__ int pos0[FTPOST], pos1[FTPOST];
    __shared__ int len0, len1;
    __shared__ unsigned char sval[NMSPOST];
    __shared__ unsigned char slab[NMSPOST];
    if (f < FTPOST) {
        sval[f] = (unsigned char)nms_val[b * NMSPOST + f];
        sval[f + FTPOST] = (unsigned char)nms_val[b * NMSPOST + f + FTPOST];
        slab[f] = (unsigned char)slabels[b * NMSPOST + f];
        slab[f + FTPOST] = (unsigned char)slabels[b * NMSPOST + f + FTPOST];
    }
    __syncthreads();
    const float* row = iou3 + (size_t)b * FTPOST * NMSPOST + (size_t)f * NMSPOST;
    int flab = flabels[b * FTPOST + f];
    int fv = ft_val[b * FTPOST + f];
#pragma unroll 2
    for (int c = 0; c < 2; ++c) {
        bool fm = fv && (flab == c);
        float v1 = -2.f, v2 = -2.f;
        int j1 = 0, j2 = 0;
        for (int j = 0; j < NMSPOST; ++j) {
            float v = (fm && sval[j] && slab[j] == c) ? row[j] : -1.f;
            if (v > v1) { v2 = v1; j2 = j1; v1 = v; j1 = j; }
            else if (v > v2) { v2 = v; j2 = j; }
        }
        if (c == 0) { m0[f] = (v2 >= MASK_TH); i0[f] = j2; }
        else        { m1[f] = (v2 >= MASK_TH); i1[f] = j2; }
    }
    __syncthreads();
    if (f == 0) {
        int c0 = 0, c1 = 0;
        for (int k = 0; k < FTPOST; ++k) {
            pos0[k] = c0; if (m0[k]) ++c0;
            pos1[k] = c1; if (m1[k]) ++c1;
        }
        len0 = c0; len1 = c1;
        // len_out = 2*FT_POST - 2*len0 - 2*len1
        out[OUT_LEN + b] = (float)(2 * FTPOST - 2 * c0 - 2 * c1);
    }
    __syncthreads();
    const float* fb = fboxes + ((size_t)b * FTPOST + f) * 7;
    float fsc = fscores[b * FTPOST + f];
    float flb = (float)(flab + 1);
#pragma unroll 2
    for (int c = 0; c < 2; ++c) {
        bool m = c == 0 ? (bool)m0[f] : (bool)m1[f];
        if (!m) continue;
        int si = c == 0 ? i0[f] : i1[f];
        int p = c == 0 ? pos0[f] : pos1[f];
        int lc = c == 0 ? len0 : len1;
        int base = c == 0 ? 0 : 2 * len0;
        const float* sf = sboxes + ((size_t)b * NMSPOST + si) * 7;
        float lwh[3], c1v[3], c2v[3];
#pragma unroll
        for (int k = 0; k < 3; ++k) {
            c1v[k] = sf[k] * LAMB + fb[k] * (1.f - LAMB);
            c2v[k] = -sf[k] * LAMB + fb[k] * (1.f + LAMB);
            lwh[k] = fmaxf(sf[3 + k], fb[3 + k]);
        }
        int ia = base + p;
        int ib = base + lc + p;
        float* rois = out + OUT_ROIS + (size_t)b * 256 * 7;
        float* scs = out + OUT_SCORES + b * 256;
        float* lbs = out + OUT_LABELS + b * 256;
        if (ia < 2 * FTPOST) {
            float* r = rois + (size_t)(NMSPOST + ia) * 7;
            r[0] = c1v[0]; r[1] = c1v[1]; r[2] = c1v[2];
            r[3] = lwh[0]; r[4] = lwh[1]; r[5] = lwh[2]; r[6] = fb[6];
            scs[NMSPOST + ia] = fsc;
            lbs[NMSPOST + ia] = flb;
        }
        if (ib < 2 * FTPOST) {
            float* r = rois + (size_t)(NMSPOST + ib) * 7;
            r[0] = c2v[0]; r[1] = c2v[1]; r[2] = c2v[2];
            r[3] = lwh[0]; r[4] = lwh[1]; r[5] = lwh[2]; r[6] = fb[6];
            scs[NMSPOST + ib] = fsc;
            lbs[NMSPOST + ib] = flb;
        }
    }
}

// =====================================================================
// launcher
// =====================================================================
extern "C" void kernel_launch(void* const* d_in, const int* in_sizes, int n_in,
                              void* d_out, int out_size, void* d_ws, size_t ws_size,
                              hipStream_t stream) {
    const float* boxes = (const float*)d_in[0];   // (B, N, 7)
    const float* cls   = (const float*)d_in[1];   // (B, N, 2)
    float* out = (float*)d_out;

    size_t off = 0;
    auto take = [&](size_t bytes) -> void* {
        void* p = (char*)d_ws + off;
        off += (bytes + 255) & ~(size_t)255;
        return p;
    };
    float* ws_scores   = (float*)take((size_t)BATCH * NPTS * 4);
    int*   ws_labels   = (int*)  take((size_t)BATCH * NPTS * 4);
    int*   ws_top_idx  = (int*)  take((size_t)BATCH * PRE * 4);
    float* ws_top_sc   = (float*)take((size_t)BATCH * PRE * 4);
    float* ws_top_bx   = (float*)take((size_t)BATCH * PRE * 7 * 4);
    float* ws_nx2      = (float*)take((size_t)BATCH * PRE * 4);
    float* ws_rad      = (float*)take((size_t)BATCH * PRE * 4);
    unsigned char* ws_gate = (unsigned char*)take((size_t)BATCH * PRE * PRE);
    float* ws_iou      = (float*)take((size_t)BATCH * PRE * PRE * 4);
    int*   ws_nms_sel  = (int*)  take((size_t)BATCH * NMSPOST * 4);
    int*   ws_nms_val  = (int*)  take((size_t)BATCH * NMSPOST * 4);
    int*   ws_ft_sel   = (int*)  take((size_t)BATCH * FTPOST * 4);
    int*   ws_ft_val   = (int*)  take((size_t)BATCH * FTPOST * 4);
    float* ws_sboxes   = (float*)take((size_t)BATCH * NMSPOST * 7 * 4);
    int*   ws_slabels  = (int*)  take((size_t)BATCH * NMSPOST * 4);
    float* ws_fboxes   = (float*)take((size_t)BATCH * FTPOST * 7 * 4);
    float* ws_fscores  = (float*)take((size_t)BATCH * FTPOST * 4);
    int*   ws_flabels  = (int*)  take((size_t)BATCH * FTPOST * 4);
    float* ws_iou3     = (float*)take((size_t)BATCH * FTPOST * NMSPOST * 4);

    k_zero<<<(out_size + 255) / 256, 256, 0, stream>>>(out, out_size);
    k_score_label<<<(BATCH * NPTS + 255) / 256, 256, 0, stream>>>(
        cls, ws_scores, ws_labels, BATCH * NPTS);
    k_topk<<<BATCH, 1024, 0, stream>>>(
        ws_scores, boxes, ws_top_idx, ws_top_sc, ws_top_bx, ws_nx2, ws_rad);
    k_gate_wmma<<<dim3(128, BATCH), 256, 0, stream>>>(
        ws_top_bx, ws_nx2, ws_rad, ws_gate);
    k_iou_bev<<<dim3((PRE * PRE + 255) / 256, BATCH), 256, 0, stream>>>(
        ws_top_bx, ws_gate, ws_iou);
    k_nms_scan<<<BATCH, PRE, 0, stream>>>(
        ws_iou, ws_nms_sel, ws_nms_val, ws_ft_sel, ws_ft_val);
    k_select<<<BATCH, 128, 0, stream>>>(
        ws_top_bx, ws_top_sc, ws_top_idx, ws_labels,
        ws_nms_sel, ws_nms_val, ws_ft_sel, out,
        ws_sboxes, ws_slabels, ws_fboxes, ws_fscores, ws_flabels);
    k_iou3d<<<dim3((FTPOST * NMSPOST + 255) / 256, BATCH), 256, 0, stream>>>(
        ws_fboxes, ws_sboxes, ws_iou3);
    k_fine<<<BATCH, FTPOST, 0, stream>>>(
        ws_iou3, ws_fboxes, ws_fscores, ws_flabels, ws_ft_val,
        ws_sboxes, ws_slabels, ws_nms_val, out);
}